// Group_84696755077884
// MI455X (gfx1250) — compile-verified
//
#include <hip/hip_runtime.h>
#include <cstddef>

// Problem constants (reference: data [32, 8192, 3] fp32)
#define NPTS   8192
#define NGROUP 512
#define KNN    32
#define GTILE  16          // centers per workgroup in knn kernel
#define CHUNK  512         // columns per LDS chunk
#define SW     516         // padded LDS row stride (floats) to avoid bank conflicts

typedef float v2f __attribute__((ext_vector_type(2)));
typedef float v8f __attribute__((ext_vector_type(8)));

// ---------------------------------------------------------------------------
// Kernel 1: Farthest point sampling. One block per batch, 1024 threads,
// 8 points per thread held in registers. Deterministic start at point 0.
// ---------------------------------------------------------------------------
__global__ __launch_bounds__(1024, 1)
void fps_kernel(const float* __restrict__ data, float* __restrict__ centers)
{
    const int b   = blockIdx.x;
    const int tid = threadIdx.x;
    const int lane = tid & 31;
    const int wave = tid >> 5;
    const float* xyz = data + (size_t)b * NPTS * 3;
    float* cent = centers + (size_t)b * NGROUP * 3;

    float px[8], py[8], pz[8], dist[8];
#pragma unroll
    for (int j = 0; j < 8; ++j) {
        const int p = tid + j * 1024;
        px[j] = xyz[p * 3 + 0];
        py[j] = xyz[p * 3 + 1];
        pz[j] = xyz[p * 3 + 2];
        dist[j] = 1.0e9f;
    }

    __shared__ float s_last[3];
    __shared__ float s_wv[32];
    __shared__ int   s_wi[32];
    __shared__ int   s_sel;

    if (tid == 0) {
        s_last[0] = px[0]; s_last[1] = py[0]; s_last[2] = pz[0];
        cent[0] = px[0]; cent[1] = py[0]; cent[2] = pz[0];   // center 0 = point 0
    }
    __syncthreads();

    for (int t = 1; t < NGROUP; ++t) {
        const float lx = s_last[0], ly = s_last[1], lz = s_last[2];

        // update min-distance, track local argmax (prefer lowest index on tie)
        float bv = -1.0f; int bi = 0;
#pragma unroll
        for (int j = 0; j < 8; ++j) {
            const float dx = px[j] - lx, dy = py[j] - ly, dz = pz[j] - lz;
            const float d = dx * dx + dy * dy + dz * dz;
            dist[j] = fminf(dist[j], d);
            if (dist[j] > bv) { bv = dist[j]; bi = tid + j * 1024; }
        }
        // wave32 argmax reduction
#pragma unroll
        for (int off = 16; off > 0; off >>= 1) {
            const float ov = __shfl_xor(bv, off, 32);
            const int   oi = __shfl_xor(bi, off, 32);
            if (ov > bv || (ov == bv && oi < bi)) { bv = ov; bi = oi; }
        }
        if (lane == 0) { s_wv[wave] = bv; s_wi[wave] = bi; }
        __syncthreads();
        // cross-wave reduction by wave 0
        if (wave == 0) {
            float v = s_wv[lane]; int i = s_wi[lane];
#pragma unroll
            for (int off = 16; off > 0; off >>= 1) {
                const float ov = __shfl_xor(v, off, 32);
                const int   oi = __shfl_xor(i, off, 32);
                if (ov > v || (ov == v && oi < i)) { v = ov; i = oi; }
            }
            if (lane == 0) s_sel = i;
        }
        __syncthreads();
        const int sel = s_sel;
        if (tid == (sel & 1023)) {
            const int j = sel >> 10;
            s_last[0] = px[j]; s_last[1] = py[j]; s_last[2] = pz[j];
            cent[t * 3 + 0] = px[j];
            cent[t * 3 + 1] = py[j];
            cent[t * 3 + 2] = pz[j];
        }
        __syncthreads();
    }
}

// ---------------------------------------------------------------------------
// Bitonic helpers (wave32, value+index keys, total order via index tiebreak)
// ---------------------------------------------------------------------------
__device__ __forceinline__ void cmp_ex(float& v, int& i, int j, int lane, bool up)
{
    const float ov = __shfl_xor(v, j, 32);
    const int   oi = __shfl_xor(i, j, 32);
    const bool lower   = (lane & j) == 0;
    const bool keepmin = (lower == up);
    const bool oless   = (ov < v) || (ov == v && oi < i);
    const bool take    = keepmin ? oless : !oless;
    if (take) { v = ov; i = oi; }
}

__device__ __forceinline__ void bitonic_sort32(float& v, int& i, int lane)
{
#pragma unroll
    for (int k = 2; k <= 32; k <<= 1) {
        const bool up = ((lane & k) == 0);
#pragma unroll
        for (int j = k >> 1; j > 0; j >>= 1)
            cmp_ex(v, i, j, lane, up);
    }
}

// ---------------------------------------------------------------------------
// Kernel 2: KNN distance matrix via V_WMMA_F32_16X16X4_F32 + wave32 bitonic
// top-32, then gather & recenter.  Grid: (NGROUP/GTILE, B), block 512 (16 waves).
//
// d2 = c^2 + x^2 - 2 c.x  as a single WMMA:
//   A row m = [-2cx, -2cy, -2cz, 1]      (16x4)
//   B col n = [ x,    y,    z,   x^2 ]   (4x16)
//   C[m][n] = c2[m]
// ---------------------------------------------------------------------------
__global__ __launch_bounds__(512, 1)
void knn_group_kernel(const float* __restrict__ data,
                      const float* __restrict__ centers,
                      float* __restrict__ out_neigh)
{
    const int b    = blockIdx.y;
    const int gt   = blockIdx.x;           // center tile (16 centers)
    const int tid  = threadIdx.x;
    const int wave = tid >> 5;
    const int lane = tid & 31;
    const int l16  = lane & 15;
    const bool hi  = (lane >= 16);         // second half-wave

    const float* xyz  = data + (size_t)b * NPTS * 3;
    const float* cent = centers + (size_t)b * NGROUP * 3 + (size_t)gt * GTILE * 3;

    __shared__ float s_cx[GTILE], s_cy[GTILE], s_cz[GTILE], s_c2[GTILE];
    __shared__ float s_d2[GTILE * SW];

    if (tid < GTILE) {
        const float x = cent[tid * 3 + 0];
        const float y = cent[tid * 3 + 1];
        const float z = cent[tid * 3 + 2];
        s_cx[tid] = x; s_cy[tid] = y; s_cz[tid] = z;
        s_c2[tid] = x * x + y * y + z * z;
    }
    __syncthreads();

    // A operand (16x4 f32): lanes 0-15 hold {K0,K1} of row M=lane,
    //                       lanes 16-31 hold {K2,K3} of row M=lane-16.
    // Branchless: every lane reads all 3 coords, then selects.
    {
        // nothing
    }
    const float acx = s_cx[l16], acy = s_cy[l16], acz = s_cz[l16];
    v2f a;
    a.x = hi ? (-2.0f * acz) : (-2.0f * acx);
    a.y = hi ? 1.0f          : (-2.0f * acy);

    // C operand (16x16 f32): VGPR v -> row v (lanes 0-15) / v+8 (lanes 16-31).
    const int mbase = hi ? 8 : 0;
    v8f cacc;
#pragma unroll
    for (int v = 0; v < 8; ++v) cacc[v] = s_c2[v + mbase];

    // running sorted-ascending top-32 across lanes; cmax = lane-31 value
    float topv = 1.0e30f;
    int   topi = 0;
    float cmax = 1.0e30f;

    for (int chunk = 0; chunk < NPTS / CHUNK; ++chunk) {
        const int cbase = chunk * CHUNK;

        // --- compute phase: each wave produces 2 of the 32 column tiles ---
#pragma unroll
        for (int r = 0; r < 2; ++r) {
            const int tcol = wave + r * 16;            // tile within chunk
            const int n = cbase + tcol * 16 + l16;     // my column's point
            const float x = xyz[n * 3 + 0];
            const float y = xyz[n * 3 + 1];
            const float z = xyz[n * 3 + 2];
            const float x2 = x * x + y * y + z * z;
            // B operand (4x16 f32): mirrors A layout (rows K0/K1 lanes 0-15,
            // rows K2/K3 lanes 16-31), column N = lane&15. Branchless select.
            v2f bb;
            bb.x = hi ? z  : x;
            bb.y = hi ? x2 : y;

            v8f d = __builtin_amdgcn_wmma_f32_16x16x4_f32(
                false, a, false, bb, (short)0, cacc, false, false);

#pragma unroll
            for (int v = 0; v < 8; ++v) {
                const int m = v + mbase;
                s_d2[m * SW + tcol * 16 + l16] = d[v];
            }
        }
        __syncthreads();

        // --- selection phase: wave g streams row g, 32 candidates at a time ---
        const int g = wave;
        for (int s = 0; s < CHUNK / 32; ++s) {
            const int col = s * 32 + lane;
            float sv = s_d2[g * SW + col];
            int   si = cbase + col;
            // wave-uniform prune: skip batch if no candidate beats current 32nd
            if (__any(sv < cmax)) {
                bitonic_sort32(sv, si, lane);          // ascending across lanes
                // merge two sorted-32 lists, keep 32 smallest
                const float rv = __shfl(topv, 31 - lane, 32);
                const int   ri = __shfl(topi, 31 - lane, 32);
                const bool take = (sv < rv) || (sv == rv && si < ri);
                float mv = take ? sv : rv;
                int   mi = take ? si : ri;
#pragma unroll
                for (int j = 16; j > 0; j >>= 1)       // bitonic merge, ascending
                    cmp_ex(mv, mi, j, lane, true);
                topv = mv; topi = mi;
                cmax = __shfl(topv, 31, 32);
            }
        }
        __syncthreads();
    }

    // --- output: lane k writes k-th nearest neighbor, recentered ---
    const int g  = wave;
    const int gg = gt * GTILE + g;
    const float ox = s_cx[g], oy = s_cy[g], oz = s_cz[g];
    const int idx = topi;
    const float nx = xyz[idx * 3 + 0] - ox;
    const float ny = xyz[idx * 3 + 1] - oy;
    const float nz = xyz[idx * 3 + 2] - oz;
    float* o = out_neigh + ((size_t)b * NGROUP * KNN + (size_t)gg * KNN + lane) * 3;
    o[0] = nx; o[1] = ny; o[2] = nz;
}

// ---------------------------------------------------------------------------
extern "C" void kernel_launch(void* const* d_in, const int* in_sizes, int n_in,
                              void* d_out, int out_size, void* d_ws, size_t ws_size,
                              hipStream_t stream)
{
    (void)n_in; (void)d_ws; (void)ws_size; (void)out_size;
    const float* data = (const float*)d_in[0];
    const int B = in_sizes[0] / (NPTS * 3);

    float* out = (float*)d_out;
    // d_out layout: neighborhood [B,512,32,3] then center [B,512,3]
    float* out_center = out + (size_t)B * NGROUP * KNN * 3;

    fps_kernel<<<dim3(B), dim3(1024), 0, stream>>>(data, out_center);
    knn_group_kernel<<<dim3(NGROUP / GTILE, B), dim3(512), 0, stream>>>(
        data, out_center, out);
}